// cluster_som_21818433864236
// MI455X (gfx1250) — compile-verified
//
#include <hip/hip_runtime.h>
#include <hip/hip_bf16.h>
#include <math.h>

// ---------------- problem constants (from reference) ----------------
#define C_N   7
#define S_N   256
#define D_N   1024
#define B_N   64
#define SD    (S_N * D_N)            // 262144
#define NBD   256                    // blocks for dist partial reduction
#define LRF   0.05f
#define DENF  157.07963267948966f    // 2*pi*sigma^2, sigma=5

typedef __attribute__((ext_vector_type(2))) float v2f;
typedef __attribute__((ext_vector_type(8))) float v8f;

// ---------------- WMMA-based wave32 reduction ----------------
// A (16x4 f32): lane L<16 supplies A[L,0] (a.x) and A[L,1] (a.y);
//               lane L>=16 supplies A[L-16,2] and A[L-16,3].
// With B = all-ones (4x16), D[m,n] = sum_k A[m,k] = p_m + p_{m+16}.
// D layout: vgpr r, lane<16 -> D[r, lane]; lane>=16 -> D[r+8, lane-16].
// So sum of a lane's 8 D values = half-sum; xor-16 partner has the other half.
__device__ __forceinline__ float wave_reduce_wmma(float p) {
    v2f a; a.x = p;    a.y = 0.0f;
    v2f b; b.x = 1.0f; b.y = 1.0f;
    v8f c = {};
    v8f d = __builtin_amdgcn_wmma_f32_16x16x4_f32(
        /*neg_a=*/false, a, /*neg_b=*/false, b,
        /*c_mod=*/(short)0, c, /*reuse_a=*/false, /*reuse_b=*/false);
    float s = d[0] + d[1] + d[2] + d[3] + d[4] + d[5] + d[6] + d[7];
    s += __shfl_xor(s, 16, 32);
    return s;   // full 32-lane sum, valid in every lane
}

// Block reduction (blockDim.x == 256 -> 8 waves). Result valid in thread 0.
__device__ __forceinline__ float block_reduce_wmma(float p, float* red) {
    float ws = wave_reduce_wmma(p);
    int lane = threadIdx.x & 31;
    int wid  = threadIdx.x >> 5;
    if (lane == 0) red[wid] = ws;
    __syncthreads();
    float t = 0.0f;
    if (threadIdx.x == 0) {
        #pragma unroll
        for (int i = 0; i < 8; ++i) t += red[i];
    }
    __syncthreads();
    return t;
}

// ---------------- kernels ----------------

// dist partials: partial[c*NBD + blk] = block-partial of sum((x - mem_c)^2)
__global__ void som_dist_partial(const float* __restrict__ x,
                                 const float* __restrict__ mem,
                                 float* __restrict__ partial) {
    __shared__ float red[8];
    float acc[C_N];
    #pragma unroll
    for (int c = 0; c < C_N; ++c) acc[c] = 0.0f;

    const int stride = NBD * 256;                       // 65536, SD/stride = 4
    for (int idx = blockIdx.x * 256 + threadIdx.x; idx < SD; idx += stride) {
        float xv = x[idx];
        #pragma unroll
        for (int c = 0; c < C_N; ++c) {
            float df = xv - mem[(size_t)c * SD + idx];
            acc[c] = fmaf(df, df, acc[c]);
        }
    }
    #pragma unroll
    for (int c = 0; c < C_N; ++c) {
        float t = block_reduce_wmma(acc[c], red);       // uniform: EXEC all ones
        if (threadIdx.x == 0) partial[c * NBD + blockIdx.x] = t;
        __syncthreads();
    }
}

// single block of 256 threads: reduce partials -> dist, softmax -> prob,
// argmin -> bmu; optionally loss = mean(prob*dist).
__global__ void som_finalize(const float* __restrict__ partial,
                             float* __restrict__ prob,
                             int* __restrict__ bmu,
                             float* __restrict__ loss /* may be null */) {
    __shared__ float red[8];
    __shared__ float dist_sh[C_N];
    #pragma unroll
    for (int c = 0; c < C_N; ++c) {
        float p = partial[c * NBD + threadIdx.x];
        float t = block_reduce_wmma(p, red);
        if (threadIdx.x == 0) dist_sh[c] = sqrtf(t);
        __syncthreads();
    }
    if (threadIdx.x == 0) {
        float mx = dist_sh[0], mn = dist_sh[0];
        int   bm = 0;
        #pragma unroll
        for (int c = 1; c < C_N; ++c) {
            mx = fmaxf(mx, dist_sh[c]);
            if (dist_sh[c] < mn) { mn = dist_sh[c]; bm = c; }
        }
        float e[C_N], se = 0.0f;
        #pragma unroll
        for (int c = 0; c < C_N; ++c) { e[c] = expf(dist_sh[c] - mx); se += e[c]; }
        float inv = 1.0f / se, l = 0.0f;
        #pragma unroll
        for (int c = 0; c < C_N; ++c) {
            float pr = e[c] * inv;
            prob[c] = pr;
            l = fmaf(pr, dist_sh[c], l);
        }
        *bmu = bm;
        if (loss) *loss = l * (1.0f / (float)C_N);
    }
}

// w[c*S + s] = LR * exp(||mem[c,s,:] - mem[bmu,s,:]|| / DEN)
// grid = C*S blocks of 256 threads; 1024/256 = 4 elems/thread (no divergence)
__global__ void som_dn(const float* __restrict__ mem,
                       const int* __restrict__ bmu_p,
                       float* __restrict__ w) {
    __shared__ float red[8];
    const int cs = blockIdx.x;                 // c*S + s
    const int c  = cs >> 8;
    const int s  = cs & (S_N - 1);
    const int bmu = *bmu_p;
    const float* row = mem + (size_t)c   * SD + (size_t)s * D_N;
    const float* cen = mem + (size_t)bmu * SD + (size_t)s * D_N;
    float acc = 0.0f;
    #pragma unroll
    for (int i = 0; i < D_N / 256; ++i) {
        int d = i * 256 + threadIdx.x;
        float df = row[d] - cen[d];
        acc = fmaf(df, df, acc);
    }
    float t = block_reduce_wmma(acc, red);
    if (threadIdx.x == 0) w[cs] = LRF * expf(sqrtf(t) / DENF);
}

// per-(s,d) update: loads all 7 centroid values -> colsum, centroid locally,
// so the read of old mem[bmu] is race-free despite in-place update.
__global__ void som_update(float* __restrict__ mem,
                           const float* __restrict__ w,
                           const int* __restrict__ bmu_p) {
    const int idx = blockIdx.x * 256 + threadIdx.x;    // 0..SD-1
    const int s   = idx >> 10;                         // D = 1024
    const int bmu = *bmu_p;
    float v[C_N];
    #pragma unroll
    for (int c = 0; c < C_N; ++c) v[c] = mem[(size_t)c * SD + idx];
    float colsum = 0.0f;
    #pragma unroll
    for (int c = 0; c < C_N; ++c) colsum += v[c];
    const float mixture = colsum * (1.0f / (float)C_N); // prob sums to 1 -> drops out
    const float delta   = mixture - v[bmu];
    #pragma unroll
    for (int c = 0; c < C_N; ++c)
        mem[(size_t)c * SD + idx] = fmaf(w[c * S_N + s], delta, v[c]);
}

// out[0 : 7*SD)          = prob[c] * colsum / 7
// out[7*SD]              = loss (written by som_finalize)
// out[7*SD+1 : 14*SD+1)  = final mem
__global__ void som_output(const float* __restrict__ mem,
                           const float* __restrict__ prob,
                           float* __restrict__ out) {
    const int idx = blockIdx.x * 256 + threadIdx.x;    // 0..SD-1
    float v[C_N];
    #pragma unroll
    for (int c = 0; c < C_N; ++c) v[c] = mem[(size_t)c * SD + idx];
    float colsum = 0.0f;
    #pragma unroll
    for (int c = 0; c < C_N; ++c) colsum += v[c];
    const float scaled = colsum * (1.0f / (float)C_N);
    float* memout = out + (size_t)C_N * SD + 1;
    #pragma unroll
    for (int c = 0; c < C_N; ++c) {
        out[(size_t)c * SD + idx]    = prob[c] * scaled;
        memout[(size_t)c * SD + idx] = v[c];
    }
}

// ---------------- host orchestration ----------------
extern "C" void kernel_launch(void* const* d_in, const int* in_sizes, int n_in,
                              void* d_out, int out_size, void* d_ws, size_t ws_size,
                              hipStream_t stream) {
    const float* x        = (const float*)d_in[0];   // [64,256,1024]
    const float* mem_init = (const float*)d_in[1];   // [7,256,1024]

    float* ws      = (float*)d_ws;
    float* mem     = ws;                             // 7*SD floats (7 MB, L2-resident)
    float* partial = mem + (size_t)C_N * SD;         // 7*NBD
    float* w       = partial + C_N * NBD;            // 7*256
    float* prob    = w + C_N * S_N;                  // 7
    int*   bmu     = (int*)(prob + C_N);             // 1

    float* out  = (float*)d_out;
    float* loss = out + (size_t)C_N * SD;            // scalar slot between output and mem

    hipMemcpyAsync(mem, mem_init, (size_t)C_N * SD * sizeof(float),
                   hipMemcpyDeviceToDevice, stream);

    for (int t = 0; t < B_N; ++t) {
        som_dist_partial<<<NBD, 256, 0, stream>>>(x + (size_t)t * SD, mem, partial);
        som_finalize   <<<1,   256, 0, stream>>>(partial, prob, bmu, nullptr);
        som_dn         <<<C_N * S_N, 256, 0, stream>>>(mem, bmu, w);
        som_update     <<<SD / 256, 256, 0, stream>>>(mem, w, bmu);
    }
    // clustering on x_last = input[63] against final mem
    som_dist_partial<<<NBD, 256, 0, stream>>>(x + (size_t)(B_N - 1) * SD, mem, partial);
    som_finalize   <<<1,   256, 0, stream>>>(partial, prob, bmu, loss);
    som_output     <<<SD / 256, 256, 0, stream>>>(mem, prob, out);
}